// LIFNeuronModel_60524679135432
// MI455X (gfx1250) — compile-verified
//
#include <hip/hip_runtime.h>
#include <stdint.h>

// ---- LIF constants (from reference) ----
#define LIF_DECAY   0.95f     // 1 - 1/TAU_M
#define LIF_BIAS   (-3.25f)   // V_REST / TAU_M
#define LIF_VREST  (-65.0f)
#define LIF_THRESH (-50.0f)

constexpr int T_STEPS = 4096;
constexpr int N_NEUR  = 16384;
constexpr int NPB     = 128;          // neurons per block == blockDim.x
constexpr int TT      = 64;           // timesteps per LDS tile
constexpr int NBI     = 4;            // input LDS buffers  (4 x 32 KB = 128 KB)
constexpr int NBO     = 2;            // output LDS buffers (2 x 32 KB =  64 KB)
constexpr int NTILES  = T_STEPS / TT; // 64
constexpr int TILE_F  = TT * NPB;     // floats per tile (8192 = 32 KB)

typedef __attribute__((ext_vector_type(4))) unsigned int v4u;
typedef __attribute__((ext_vector_type(8))) int          v8i;
typedef __attribute__((ext_vector_type(4))) int          v4i;

// Build the D# for a TT x NPB fp32 tile of the (T_STEPS, N_NEUR) row-major
// tensor, tile origin at gaddr, LDS byte offset lds_off.
// group0: count=1 | lds_addr | global_addr | type=2
// group1: data_size=4B, tensor_dim0=N, tensor_dim1=T, tile_dim0=NPB,
//         tile_dim1=TT, tensor_dim0_stride=N. Groups 2/3 + extra: zero (2D).
__device__ __forceinline__ void tdm_build(v4u& g0, v8i& g1,
                                          const void* gaddr, unsigned lds_off) {
  uint64_t ga = (uint64_t)(uintptr_t)gaddr;
  g0[0] = 1u;
  g0[1] = lds_off;
  g0[2] = (unsigned)(ga & 0xFFFFFFFFu);
  g0[3] = (unsigned)((ga >> 32) & 0x01FFFFFFu) | (2u << 30);

  g1[0] = (int)(2u << 16);                               // data_size = 4B
  g1[1] = (int)(((unsigned)N_NEUR & 0xFFFFu) << 16);     // tensor_dim0 lo16
  g1[2] = (int)((((unsigned)N_NEUR >> 16) & 0xFFFFu)
        |  (((unsigned)T_STEPS & 0xFFFFu) << 16));       // dim0 hi | dim1 lo
  g1[3] = (int)((((unsigned)T_STEPS >> 16) & 0xFFFFu)
        |  ((unsigned)NPB << 16));                       // dim1 hi | tile_dim0
  g1[4] = (int)(unsigned)TT;                             // tile_dim1 (tile_dim2=0)
  g1[5] = N_NEUR;                                        // tensor_dim0_stride
  g1[6] = 0;
  g1[7] = 0;
}

__device__ __forceinline__ void tdm_load_tile(const float* gsrc, unsigned lds_off) {
  v4u g0; v8i g1;
  tdm_build(g0, g1, gsrc, lds_off);
  v4i gz  = {0, 0, 0, 0};
  v8i gz8 = {0, 0, 0, 0, 0, 0, 0, 0};
  __builtin_amdgcn_tensor_load_to_lds(g0, g1, gz, gz, gz8, 0);
}

__device__ __forceinline__ void tdm_store_tile(float* gdst, unsigned lds_off) {
  v4u g0; v8i g1;
  tdm_build(g0, g1, gdst, lds_off);
  v4i gz  = {0, 0, 0, 0};
  v8i gz8 = {0, 0, 0, 0, 0, 0, 0, 0};
  __builtin_amdgcn_tensor_store_from_lds(g0, g1, gz, gz, gz8, 0);
}

__global__ __launch_bounds__(NPB, 1)
void lif_scan_tdm_kernel(const float* __restrict__ win,
                         const float* __restrict__ vinit,
                         float* __restrict__ out) {
  __shared__ float s_in [NBI * TILE_F];   // 128 KB input staging
  __shared__ float s_out[NBO * TILE_F];   //  64 KB output staging

  const int tid   = threadIdx.x;
  const int wid   = tid >> 5;             // wave id (wave32)
  const int nbase = blockIdx.x * NPB;
  const int n     = nbase + tid;

  float v = vinit[n];

  const unsigned lds_in  = (unsigned)(uintptr_t)s_in;   // LDS byte offsets
  const unsigned lds_out = (unsigned)(uintptr_t)s_out;

  // Wave 0 primes the input pipeline: NBI tiles in flight via TDM.
  if (wid == 0) {
    for (int i = 0; i < NBI; ++i) {
      tdm_load_tile(win + (size_t)i * TT * N_NEUR + nbase,
                    lds_in + (unsigned)(i * TILE_F * 4));
    }
  }

  for (int tile = 0; tile < NTILES; ++tile) {
    const int bi = tile & (NBI - 1);
    const int bo = tile & (NBO - 1);

    // Wave 0: loads complete in order; <= NBI-1 outstanding => this tile's
    // input landed in LDS.  Wave 1: <= NBO-1 outstanding stores => the
    // out-buffer we are about to overwrite has been drained by the TDM.
    if (wid == 0) __builtin_amdgcn_s_wait_tensorcnt(NBI - 1);
    if (wid == 1) __builtin_amdgcn_s_wait_tensorcnt(NBO - 1);
    __syncthreads();   // A: in-buf valid, out-buf free, for all waves

    const float* src = s_in  + bi * TILE_F + tid;
    float*       dst = s_out + bo * TILE_F + tid;
#pragma unroll 16
    for (int tt = 0; tt < TT; ++tt) {
      float inp = src[tt * NPB];                  // conflict-free ds_load
      v = fmaf(v, LIF_DECAY, LIF_BIAS) + inp;     // leaky integrate
      bool fire = (v >= LIF_THRESH);
      dst[tt * NPB] = fire ? 1.0f : 0.0f;         // spike row -> LDS out
      v = fire ? LIF_VREST : v;                   // reset
    }

    __syncthreads();   // B: compute done; in-buf consumed, out-buf complete

    // Wave 0 refills the freed input buffer NBI tiles ahead.
    const int nxt = tile + NBI;
    if (wid == 0 && nxt < NTILES) {
      tdm_load_tile(win + (size_t)nxt * TT * N_NEUR + nbase,
                    lds_in + (unsigned)(bi * TILE_F * 4));
    }
    // Wave 1 streams this tile's spikes out via TDM.
    if (wid == 1) {
      tdm_store_tile(out + (size_t)tile * TT * N_NEUR + nbase,
                     lds_out + (unsigned)(bo * TILE_F * 4));
    }
  }
  // Outstanding TDM stores are retired by S_ENDPGM's implicit wait-idle.
}

extern "C" void kernel_launch(void* const* d_in, const int* in_sizes, int n_in,
                              void* d_out, int out_size, void* d_ws, size_t ws_size,
                              hipStream_t stream) {
  const float* weighted_input     = (const float*)d_in[0];  // (T, N) fp32
  const float* membrane_potential = (const float*)d_in[1];  // (N,)  fp32
  float*       spikes             = (float*)d_out;          // (T, N) fp32

  (void)in_sizes; (void)n_in; (void)out_size; (void)d_ws; (void)ws_size;

  dim3 grid(N_NEUR / NPB);   // 128 blocks
  dim3 block(NPB);           // 128 threads = 4 waves (wave32)
  lif_scan_tdm_kernel<<<grid, block, 0, stream>>>(weighted_input, membrane_potential, spikes);
}